// SimpleAverageSelfAttention_40922448396698
// MI455X (gfx1250) — compile-verified
//
#include <hip/hip_runtime.h>

// SimpleAverageSelfAttention on MI455X (gfx1250):
//   out[b,t,c] = (1/(t+1)) * sum_{s<=t} x[b,s,c]
// Two-pass chunked prefix scan. Intra-tile 16x16 cumsum is computed as
// D = L16 @ Xtile via 4 chained V_WMMA_F32_16X16X4_F32 (exact f32; L is a
// 0/1 lower-triangular matrix). The 1/(t+1) softmax weight is applied as a
// v_rcp_f32 multiply (matches the reference's "normalize then multiply"
// structure; <=1 ulp on the weight).
//
// B=16, T=4096, C=256. Traffic ~192MB -> ~8us at 23.3 TB/s (memory bound).

#define B_  16
#define T_  4096
#define C_  256
#define CHUNK_T 256
#define NCHUNK  (T_ / CHUNK_T)   // 16
#define SUBT    16
#define NSUB    (CHUNK_T / SUBT) // 16

typedef float v2f __attribute__((ext_vector_type(2)));
typedef float v8f __attribute__((ext_vector_type(8)));

// ---------------------------------------------------------------------------
// Pass 1: per-(b, chunk, channel) sums of 256 timesteps.
// Grid: B*NCHUNK*(C/64) = 1024 blocks x 256 threads. Fully coalesced rows.
// ---------------------------------------------------------------------------
__global__ void __launch_bounds__(256)
chunk_sums_kernel(const float* __restrict__ x, float* __restrict__ partials) {
    int blk   = blockIdx.x;
    int ctile = blk & 3;                 // C/64 = 4
    int chunk = (blk >> 2) & (NCHUNK - 1);
    int b     = blk >> 6;
    int tid   = threadIdx.x;
    int c     = ctile * 64 + (tid & 63);
    int g     = tid >> 6;                // 4 t-groups of 64

    const float* p = x + ((size_t)b * T_ + (size_t)chunk * CHUNK_T + g * 64) * C_ + c;
    float s = 0.0f;
    #pragma unroll 8
    for (int i = 0; i < 64; ++i) s += p[(size_t)i * C_];

    __shared__ float red[256];
    red[tid] = s;
    __syncthreads();
    if (g == 0) {
        float tot = red[tid] + red[tid + 64] + red[tid + 128] + red[tid + 192];
        partials[((size_t)b * NCHUNK + chunk) * C_ + c] = tot;
    }
}

// ---------------------------------------------------------------------------
// Pass 2: one wave (32 lanes) per (b, chunk, 16-channel tile).
// Running offset = sum of preceding chunk totals; then 16 sub-tiles of 16t,
// each scanned with 4 chained V_WMMA_F32_16X16X4_F32 (D = L @ X + C).
// C/D layout: lane holds column n = lane&15; VGPR r holds row m = r + 8*(lane>>4).
// A layout (16x4 f32): lane -> M = lane&15; VGPR v -> K = v + 2*(lane>>4).
// B layout (4x16 f32): lane -> N = lane&15; VGPR v -> K = v + 2*(lane>>4).
// ---------------------------------------------------------------------------
__global__ void __launch_bounds__(32)
scan_kernel(const float* __restrict__ x,
            const float* __restrict__ partials,
            float* __restrict__ out) {
    int blk   = blockIdx.x;
    int ctile = blk & 15;                 // C/16 = 16
    int chunk = (blk >> 4) & (NCHUNK - 1);
    int b     = blk >> 8;

    int lane = threadIdx.x;               // 0..31, wave32
    int n    = lane & 15;                 // column (channel within tile)
    int half = lane >> 4;                 // half-wave select
    int c0   = ctile * 16;

    // Exclusive running offset for this chunk (<=15 tiny loads per lane;
    // lanes 16..31 duplicate lanes 0..15 -> exactly the C-matrix broadcast).
    float off = 0.0f;
    const float* pp = partials + (size_t)b * NCHUNK * C_ + c0 + n;
    for (int j = 0; j < chunk; ++j) off += pp[(size_t)j * C_];

    v8f acc;
    #pragma unroll
    for (int r = 0; r < 8; ++r) acc[r] = off;

    const float* xb = x   + ((size_t)b * T_ + (size_t)chunk * CHUNK_T) * C_;
    float*       ob = out + ((size_t)b * T_ + (size_t)chunk * CHUNK_T) * C_;

    for (int sub = 0; sub < NSUB; ++sub) {
        int tbase = sub * SUBT;

        // Load X sub-tile (16t x 16c) directly in B-matrix layout.
        v2f Bm[4];
        #pragma unroll
        for (int j = 0; j < 4; ++j) {
            #pragma unroll
            for (int v = 0; v < 2; ++v) {
                int t = tbase + 4 * j + v + 2 * half;   // K index within tile
                Bm[j][v] = xb[(size_t)t * C_ + c0 + n];
            }
        }

        // D = (sum_j L_j @ B_j) + C   with L_j[m,k] = ((4j+k) <= m)
        v8f d = acc;
        #pragma unroll
        for (int j = 0; j < 4; ++j) {
            v2f Am;
            #pragma unroll
            for (int v = 0; v < 2; ++v) {
                int k = 4 * j + v + 2 * half;
                Am[v] = (k <= n) ? 1.0f : 0.0f;         // A's M index == lane&15
            }
            d = __builtin_amdgcn_wmma_f32_16x16x4_f32(
                    /*neg_a=*/false, Am, /*neg_b=*/false, Bm[j],
                    /*c_mod=*/(short)0, d, /*reuse_a=*/false, /*reuse_b=*/false);
        }

        // Apply the uniform-attention weight w = 1/(t+1) via v_rcp_f32 and
        // store (coalesced 64B rows).
        #pragma unroll
        for (int r = 0; r < 8; ++r) {
            int m  = r + 8 * half;
            int t  = tbase + m;
            int tg = chunk * CHUNK_T + t;
            float w = __builtin_amdgcn_rcpf((float)(tg + 1));
            ob[(size_t)t * C_ + c0 + n] = d[r] * w;
        }

        // New running prefix = row 15 of D (VGPR 7, lanes 16..31); broadcast
        // per-column to both half-waves.
        float row15 = __shfl(d[7], 16 + n, 32);
        #pragma unroll
        for (int r = 0; r < 8; ++r) acc[r] = row15;
    }
}

// ---------------------------------------------------------------------------
extern "C" void kernel_launch(void* const* d_in, const int* in_sizes, int n_in,
                              void* d_out, int out_size, void* d_ws, size_t ws_size,
                              hipStream_t stream) {
    (void)in_sizes; (void)n_in; (void)out_size; (void)ws_size;
    const float* x   = (const float*)d_in[0];
    float*       out = (float*)d_out;
    float*       partials = (float*)d_ws;   // B*NCHUNK*C floats = 256 KB scratch

    chunk_sums_kernel<<<dim3(B_ * NCHUNK * (C_ / 64)), dim3(256), 0, stream>>>(x, partials);
    scan_kernel<<<dim3(B_ * NCHUNK * (C_ / 16)), dim3(32), 0, stream>>>(x, partials, out);
}